// SelfLockingControlBinder_63599875719395
// MI455X (gfx1250) — compile-verified
//
#include <hip/hip_runtime.h>

typedef __attribute__((ext_vector_type(2))) float v2f;
typedef __attribute__((ext_vector_type(8))) float v8f;

#define N      30
#define NITER  20
#define STRIDE 33                      // odd LDS row stride -> conflict-free on 64 banks
#define LOG2E  1.44269504088896340736f

__device__ __forceinline__ float fast_exp2(float x) {
#if __has_builtin(__builtin_amdgcn_exp2f)
    return __builtin_amdgcn_exp2f(x);          // raw v_exp_f32 (base-2)
#else
    return exp2f(x);
#endif
}
__device__ __forceinline__ float fast_log2(float x) {
#if __has_builtin(__builtin_amdgcn_logf)
    return __builtin_amdgcn_logf(x);           // raw v_log_f32 (base-2)
#else
    return __log2f(x);
#endif
}

// One logsumexp2 normalization of the 30 register-resident values.
// 4-way partial max/sum to break the dependence chains; TRANS (v_exp) is the
// critical path and co-issues against the VALU max/add/sub stream.
__device__ __forceinline__ void lse2_normalize(float* __restrict__ r) {
    float m0 = r[0], m1 = r[1], m2 = r[2], m3 = r[3];
#pragma unroll
    for (int j = 4; j < 28; j += 4) {
        m0 = fmaxf(m0, r[j]);     m1 = fmaxf(m1, r[j + 1]);
        m2 = fmaxf(m2, r[j + 2]); m3 = fmaxf(m3, r[j + 3]);
    }
    m0 = fmaxf(m0, r[28]); m1 = fmaxf(m1, r[29]);
    const float m = fmaxf(fmaxf(m0, m1), fmaxf(m2, m3));

    float s0 = 0.0f, s1 = 0.0f, s2 = 0.0f, s3 = 0.0f;
#pragma unroll
    for (int j = 0; j < 28; j += 4) {
        s0 += fast_exp2(r[j] - m);     s1 += fast_exp2(r[j + 1] - m);
        s2 += fast_exp2(r[j + 2] - m); s3 += fast_exp2(r[j + 3] - m);
    }
    s0 += fast_exp2(r[28] - m); s1 += fast_exp2(r[29] - m);

    const float lse = m + fast_log2((s0 + s1) + (s2 + s3));
#pragma unroll
    for (int j = 0; j < N; ++j) r[j] -= lse;
}

__global__ __launch_bounds__(32)
void sinkhorn30_wmma_kernel(const float* __restrict__ scores,
                            const float* __restrict__ action,
                            float* __restrict__ out)
{
    __shared__ float lm[32 * STRIDE];  // 32x33 padded f32 tile (4224 B)
    __shared__ float sa[32];           // zero-padded action vector

    const int       lane = threadIdx.x;   // 0..31 (wave32)
    const long long b    = blockIdx.x;    // problem id

    // ---- load row `lane`, pre-scaled into log2 domain (lse is scale-equivariant) ----
    float r[N];
    const float* srow = scores + b * (long long)(N * N) + (long long)lane * N;
    if (lane < N) {
        const v2f* s2 = (const v2f*)srow;  // rows are 8B-aligned (120B apart)
#pragma unroll
        for (int j = 0; j < N / 2; ++j) {
            v2f v = s2[j];
            r[2 * j]     = v.x * LOG2E;
            r[2 * j + 1] = v.y * LOG2E;
        }
    } else {
#pragma unroll
        for (int j = 0; j < N; ++j) r[j] = 0.0f;  // dummy rows, confined to pad
    }
    sa[lane] = (lane < N) ? action[b * N + lane] : 0.0f;

    // ---- 20 alternating row/column logsumexp2 normalizations ----
#pragma unroll 1
    for (int it = 0; it < NITER; ++it) {
        lse2_normalize(r);                              // row pass (lane = row)
        // transpose via LDS: lane becomes column owner
#pragma unroll
        for (int j = 0; j < N; ++j) lm[j * STRIDE + lane] = r[j];
        __syncthreads();
#pragma unroll
        for (int j = 0; j < N; ++j) r[j] = lm[lane * STRIDE + j];
        __syncthreads();
        lse2_normalize(r);                              // column pass (lane = column)
        if (it != NITER - 1) {
            // transpose back: lane = row owner for the next iteration
#pragma unroll
            for (int j = 0; j < N; ++j) lm[j * STRIDE + lane] = r[j];
            __syncthreads();
#pragma unroll
            for (int j = 0; j < N; ++j) r[j] = lm[lane * STRIDE + j];
            __syncthreads();
        }
    }

    // ---- lane = column owner: exp2 and store transposed == row-major P in LDS ----
#pragma unroll
    for (int j = 0; j < N; ++j) lm[j * STRIDE + lane] = fast_exp2(r[j]);
    // zero the 32x32 halo (later wave instructions win over earlier garbage stores)
    lm[lane * STRIDE + 30] = 0.0f;     // cols 30,31 of every row
    lm[lane * STRIDE + 31] = 0.0f;
    lm[30 * STRIDE + lane] = 0.0f;     // rows 30,31, every col
    lm[31 * STRIDE + lane] = 0.0f;
    __syncthreads();

    // ---- out = P * action via V_WMMA_F32_16X16X4_F32 ----
    // A 16x4 f32: lanes 0-15 -> (M=lane, K=0/1); lanes 16-31 -> (M=lane-16, K=2/3).
    // B 4x16: action[k] replicated across all 16 columns (zero-padded k>=30).
    const int half = (lane < 16) ? 0 : 2;
    const int mrow = (lane & 15);
    v8f c0 = {};   // rows 0..15
    v8f c1 = {};   // rows 16..31
#pragma unroll
    for (int t = 0; t < 8; ++t) {
        const int k0 = 4 * t + half;
        v2f bf; bf.x = sa[k0];                        bf.y = sa[k0 + 1];
        v2f a0; a0.x = lm[mrow * STRIDE + k0];        a0.y = lm[mrow * STRIDE + k0 + 1];
        v2f a1; a1.x = lm[(16 + mrow) * STRIDE + k0]; a1.y = lm[(16 + mrow) * STRIDE + k0 + 1];
        c0 = __builtin_amdgcn_wmma_f32_16x16x4_f32(false, a0, false, bf, (short)0, c0, false, false);
        c1 = __builtin_amdgcn_wmma_f32_16x16x4_f32(false, a1, false, bf, (short)0, c1, false, false);
    }

    // C/D layout: VGPR q -> lanes 0-15: D[q][n]; lanes 16-31: D[8+q][n-16].
    // B is column-replicated, so lanes 0 and 16 hold the full result column.
    float* orow = out + b * N;
    if (lane == 0) {
#pragma unroll
        for (int q = 0; q < 8; ++q) orow[q]      = c0[q];   // rows 0..7
#pragma unroll
        for (int q = 0; q < 8; ++q) orow[16 + q] = c1[q];   // rows 16..23
    } else if (lane == 16) {
#pragma unroll
        for (int q = 0; q < 8; ++q) orow[8 + q]  = c0[q];   // rows 8..15
#pragma unroll
        for (int q = 0; q < 6; ++q) orow[24 + q] = c1[q];   // rows 24..29
    }
}

extern "C" void kernel_launch(void* const* d_in, const int* in_sizes, int n_in,
                              void* d_out, int out_size, void* d_ws, size_t ws_size,
                              hipStream_t stream) {
    const float* scores = (const float*)d_in[0];
    const float* action = (const float*)d_in[1];
    float*       out    = (float*)d_out;

    const int B = in_sizes[1] / N;     // action is [B, 30]
    dim3 grid((unsigned)B), block(32);
    hipLaunchKernelGGL(sinkhorn30_wmma_kernel, grid, block, 0, stream,
                       scores, action, out);
}